// WaveConv3d_19224273616928
// MI455X (gfx1250) — compile-verified
//
#include <hip/hip_runtime.h>

typedef float v2f __attribute__((ext_vector_type(2)));
typedef float v8f __attribute__((ext_vector_type(8)));

// db4 analysis filters (pywt convention, ascending index) and QMF high-pass
__constant__ float DEC_LO[8] = {
    -0.010597401784997278f,  0.032883011666982945f,  0.030841381835986965f,
    -0.18703481171888114f,  -0.02798376941698385f,   0.6308807679295904f,
     0.7148465705525415f,    0.23037781330885523f};
__constant__ float DEC_HI[8] = {
     0.23037781330885523f,  -0.7148465705525415f,    0.6308807679295904f,
     0.02798376941698385f,  -0.18703481171888114f,  -0.030841381835986965f,
     0.032883011666982945f,  0.010597401784997278f};

// ---------------------------------------------------------------------------
// Forward DWT along one axis viewed as (outer, N, INNER) -> (outer, K, INNER)
// out_lo[j] = sum_t x[wrap(2j+t-8)] * DEC_LO[7-t]   (periodic pad 8, stride 2)
// N/INNER/K are compile-time so index div/mod strength-reduces; HAS_HI
// selects whether the high-pass band is produced (level-1 details are dead).
// ---------------------------------------------------------------------------
template <int N, int INNER, int K, bool HAS_HI>
__global__ __launch_bounds__(256) void dwt_fwd(
    const float* __restrict__ x, float* __restrict__ lo, float* __restrict__ hi,
    unsigned total)
{
    unsigned tid = blockIdx.x * 256u + threadIdx.x;
    if (tid >= total) return;
    unsigned c = tid % (unsigned)INNER;
    unsigned j = (tid / (unsigned)INNER) % (unsigned)K;
    unsigned o = tid / (unsigned)(INNER * K);
    const float* xb = x + (size_t)o * (unsigned)(N * INNER) + c;

    float alo = 0.f, ahi = 0.f;
#pragma unroll
    for (int t = 0; t < 8; ++t) {
        int p = 2 * (int)j + t - 8;           // periodic wrap into [0,N)
        p += (p < 0) ? N : 0;
        p -= (p >= N) ? N : 0;
        float v = xb[(size_t)p * INNER];
        alo += v * DEC_LO[7 - t];
        if (HAS_HI) ahi += v * DEC_HI[7 - t];
    }
    size_t oi = ((size_t)o * K + j) * INNER + c;
    lo[oi] = alo;
    if (HAS_HI) hi[oi] = ahi;
}

// ---------------------------------------------------------------------------
// Inverse DWT along one axis: (outer,K,INNER)x2 -> (outer,N,INNER)
// out[j] = sum_{t == (j+1) mod 2} lo[(j+1+t)/2]*DEC_LO[t] + hi[...]*DEC_HI[t]
// (upsample-by-2 synthesis conv, crop pad==8; matches reference exactly)
// HAS_HI=false => detail band identically zero (level-1 reconstruction).
// ---------------------------------------------------------------------------
template <int K, int INNER, int N, bool HAS_HI>
__global__ __launch_bounds__(256) void dwt_inv(
    const float* __restrict__ lo, const float* __restrict__ hi,
    float* __restrict__ out, unsigned total)
{
    unsigned tid = blockIdx.x * 256u + threadIdx.x;
    if (tid >= total) return;
    unsigned c = tid % (unsigned)INNER;
    unsigned j = (tid / (unsigned)INNER) % (unsigned)N;
    unsigned o = tid / (unsigned)(INNER * N);
    const float* lob = lo + (size_t)o * (unsigned)(K * INNER) + c;
    const float* hib = hi + (size_t)o * (unsigned)(K * INNER) + c;

    int t0 = (int)((j + 1u) & 1u);
    float acc = 0.f;
#pragma unroll
    for (int s = 0; s < 4; ++s) {
        int t = t0 + 2 * s;
        int m = (int)(j + 1u + (unsigned)t) >> 1;
        if (m < K) {
            acc += lob[(size_t)m * INNER] * DEC_LO[t];
            if (HAS_HI) acc += hib[(size_t)m * INNER] * DEC_HI[t];
        }
    }
    out[((size_t)o * N + j) * INNER + c] = acc;
}

// ---------------------------------------------------------------------------
// Channel mix in wavelet domain via V_WMMA_F32_16X16X4_F32.
// For each (band, spatial loc s in 23^3, N-tile t): D = sum_k A(16x4)B(4x16)+C
//   A[m,k] = band[b=m, i=k, s]     B[k,o] = w_band[i=k, o, s]
// Padding rows/cols (m>=8, o>=40) are never stored, and in D=A*B+C output
// row m / col o depend only on A row m / B col o — so instead of predicated
// zero-fill we CLAMP the addresses (mrow&7, min(ocol,39)) and load
// unconditionally: branch-free, no EXEC round-trips, loads batch freely
// ahead of the WMMA chain. 10 chained WMMAs cover K=40. 292,008 waves =
// 36,501 blocks x 8 waves exactly, so EXEC is all-ones at every WMMA.
// VGPR layouts per CDNA5 ISA 7.12.2: A/B 2 VGPRs (even K in v0, odd in v1;
// lanes 16-31 carry K+2), C/D v8f rows r + 8*(lane>=16), col = lane&15.
// ---------------------------------------------------------------------------
__global__ __launch_bounds__(256) void mix_wmma(
    const float* __restrict__ bands, float* __restrict__ outb,
    const float* __restrict__ w0, const float* __restrict__ w1,
    const float* __restrict__ w2, const float* __restrict__ w3,
    const float* __restrict__ w4, const float* __restrict__ w5,
    const float* __restrict__ w6, const float* __restrict__ w7)
{
    const int lane = threadIdx.x & 31;
    // wave-uniform: force scalar so div/mod and pointer select go to SALU
    const unsigned wid =
        __builtin_amdgcn_readfirstlane(blockIdx.x * 8u + (threadIdx.x >> 5));
    const int tile = (int)(wid % 3u);
    const unsigned rem = wid / 3u;
    const int loc  = (int)(rem % 12167u);
    const int band = (int)(rem / 12167u);

    const float* w;
    switch (band) {
        case 0: w = w0; break; case 1: w = w1; break;
        case 2: w = w2; break; case 3: w = w3; break;
        case 4: w = w4; break; case 5: w = w5; break;
        case 6: w = w6; break; default: w = w7; break;
    }
    const float* in = bands + (size_t)band * 3893440u;
    float* out      = outb  + (size_t)band * 3893440u;

    const int mrow = lane & 15;       // A row / B col inside the 16-wide tile
    const int half = lane >> 4;       // K-pair select (ISA layout)
    const int ocol = tile * 16 + (lane & 15);
    const int mrowc = mrow & 7;                   // clamp: dup rows, discarded
    const int ocolc = (ocol < 40) ? ocol : 39;    // clamp: dup col, discarded

    // Per-lane bases; all 20 A loads fit in the 24-bit immediate offset.
    const float* abase = in + (size_t)(mrowc * 40) * 12167u + (unsigned)loc;
    const float* bbase = w  + (size_t)ocolc * 12167u + (unsigned)loc;

    v8f acc = {};
#pragma unroll
    for (int kt = 0; kt < 10; ++kt) {
        const int k0 = kt * 4 + half * 2;
        v2f a, b;
        a.x = abase[(size_t)k0 * 12167u];
        a.y = abase[(size_t)(k0 + 1) * 12167u];
        b.x = bbase[(size_t)k0 * 486680u];        // stride Co*23^3 between k
        b.y = bbase[(size_t)(k0 + 1) * 486680u];
        acc = __builtin_amdgcn_wmma_f32_16x16x4_f32(
            /*neg_a=*/false, a, /*neg_b=*/false, b,
            /*c_mod=*/(short)0, acc, /*reuse_a=*/false, /*reuse_b=*/false);
    }
    if (half == 0 && ocol < 40) {     // rows 0..7 live in lanes 0..15
#pragma unroll
        for (int r = 0; r < 8; ++r)
            out[(size_t)((r * 40 + ocol) * 12167 + loc)] = acc[r];
    }
}

// ---------------------------------------------------------------------------
// Host-side orchestration
// ---------------------------------------------------------------------------
template <int N, int INNER, int K, bool HAS_HI>
static inline void launch_fwd(const float* x, float* lo, float* hi,
                              unsigned outer, hipStream_t s) {
    unsigned total = outer * (unsigned)(K * INNER);
    hipLaunchKernelGGL((dwt_fwd<N, INNER, K, HAS_HI>),
                       dim3((total + 255u) / 256u), dim3(256), 0, s,
                       x, lo, hi, total);
}
template <int K, int INNER, int N, bool HAS_HI>
static inline void launch_inv(const float* lo, const float* hi, float* out,
                              unsigned outer, hipStream_t s) {
    unsigned total = outer * (unsigned)(N * INNER);
    hipLaunchKernelGGL((dwt_inv<K, INNER, N, HAS_HI>),
                       dim3((total + 255u) / 256u), dim3(256), 0, s,
                       lo, hi, out, total);
}

extern "C" void kernel_launch(void* const* d_in, const int* in_sizes, int n_in,
                              void* d_out, int out_size, void* d_ws, size_t ws_size,
                              hipStream_t stream) {
    const float* x = (const float*)d_in[0];
    const float* w[8];
    for (int i = 0; i < 8; ++i) w[i] = (const float*)d_in[1 + i];
    float* out = (float*)d_out;

    // Scratch layout (floats). BC = B*Ci = 320; band = 320*23^3 = 3,893,440.
    float* ws = (float*)d_ws;
    float* A = ws;                       // 48,496,640  (320,37,64,64) & reuse
    float* B = A + 48496640u;            // 28,037,120  (320,37,37,64) & reuse
    float* C = B + 28037120u;            // 16,208,960  (320,37,37,37) & reuse
    float* D = C + 16208960u;            // 31,147,520  8 level-2 bands
    float* E = D + 31147520u;            // 31,147,520  8 mixed bands
    const unsigned L2X = 10075840u;      // 320*23*37*37
    const unsigned L2Y = 6263360u;       // 320*23*23*37
    const unsigned BSZ = 3893440u;       // 320*23*23*23

    // ---- forward level 1: approx (low-pass) branch only -------------------
    launch_fwd<64, 4096, 37, false>(x, A, nullptr, 320u,         stream); // X
    launch_fwd<64, 64,   37, false>(A, B, nullptr, 320u * 37u,   stream); // Y
    launch_fwd<64, 1,    37, false>(B, C, nullptr, 320u * 1369u, stream); // Z -> a1

    // ---- forward level 2: full 8-band decomposition -----------------------
    launch_fwd<37, 1369, 23, true>(C,          A,            A + L2X,      320u,        stream); // X: a2,d2
    launch_fwd<37, 37,   23, true>(A,          B,            B + L2Y,      320u * 23u,  stream); // Y: aa,ad
    launch_fwd<37, 37,   23, true>(A + L2X,    B + 2u * L2Y, B + 3u * L2Y, 320u * 23u,  stream); // Y: da,dd
    launch_fwd<37, 1,    23, true>(B,          D,            D + BSZ,      320u * 529u, stream); // Z: aaa,aad
    launch_fwd<37, 1,    23, true>(B + L2Y,    D + 2u * BSZ, D + 3u * BSZ, 320u * 529u, stream); // Z: ada,add
    launch_fwd<37, 1,    23, true>(B + 2u*L2Y, D + 4u * BSZ, D + 5u * BSZ, 320u * 529u, stream); // Z: daa,dad
    launch_fwd<37, 1,    23, true>(B + 3u*L2Y, D + 6u * BSZ, D + 7u * BSZ, 320u * 529u, stream); // Z: dda,ddd

    // ---- channel mix on all 8 bands via f32 WMMA --------------------------
    // 8 bands * 12167 locs * 3 N-tiles = 292,008 waves = 36,501 blocks * 8
    hipLaunchKernelGGL(mix_wmma, dim3(36501), dim3(256), 0, stream,
                       D, E, w[0], w[1], w[2], w[3], w[4], w[5], w[6], w[7]);

    // ---- inverse level 2 --------------------------------------------------
    launch_inv<23, 1,    37, true>(E,            E + BSZ,      A,            320u * 529u, stream); // Z
    launch_inv<23, 1,    37, true>(E + 2u * BSZ, E + 3u * BSZ, A + L2Y,      320u * 529u, stream);
    launch_inv<23, 1,    37, true>(E + 4u * BSZ, E + 5u * BSZ, A + 2u * L2Y, 320u * 529u, stream);
    launch_inv<23, 1,    37, true>(E + 6u * BSZ, E + 7u * BSZ, A + 3u * L2Y, 320u * 529u, stream);
    launch_inv<23, 37,   37, true>(A,            A + L2Y,      B,            320u * 23u,  stream); // Y
    launch_inv<23, 37,   37, true>(A + 2u * L2Y, A + 3u * L2Y, B + L2X,      320u * 23u,  stream);
    launch_inv<23, 1369, 37, true>(B,            B + L2X,      C,            320u,        stream); // X -> rec2

    // ---- inverse level 1 (detail bands identically zero) ------------------
    launch_inv<37, 1,    64, false>(C, nullptr, D,   320u * 1369u, stream); // Z
    launch_inv<37, 64,   64, false>(D, nullptr, A,   320u * 37u,   stream); // Y
    launch_inv<37, 4096, 64, false>(A, nullptr, out, 320u,         stream); // X
}